// Impedance_64776696758819
// MI455X (gfx1250) — compile-verified
//
#include <hip/hip_runtime.h>
#include <stdint.h>

#define TPB 256
#define WAVES_PER_BLOCK (TPB / 32)
// Per wave-region: 16 async b128 stores x 32 lanes x 16B = 8192 B = 512 chunks.
#define CHUNKS_PER_WAVE 512ull

// ---------------------------------------------------------------------------
// Kernel 1: stream zeros over the whole N*N output using the gfx1250 async
// LDS->global store path (GLOBAL_STORE_ASYNC_FROM_LDS_B128, ASYNCcnt).
// Each wave zeroes nothing in VGPRs: zeros are sourced from a 512B LDS
// region. Each wave owns one contiguous 8KB slice and issues 16 clause'd
// async stores with immediate offsets -- no address math in the hot path.
// NT hint: 604MB output >> 192MB L2, so bypass temporal caching.
// ---------------------------------------------------------------------------
__global__ void zero_fill_async_kernel(float* __restrict__ out,
                                       unsigned long long n_chunks,   // 16B chunks
                                       unsigned int n_tail,           // leftover floats
                                       unsigned long long total_elems) {
  __shared__ __align__(64) float zbuf[128];  // 512 bytes of zeros
  const unsigned tid = threadIdx.x;
  if (tid < 128u) zbuf[tid] = 0.0f;
  __syncthreads();

  const unsigned lane = tid & 31u;
  // Low 32 bits of a generic pointer to LDS == LDS byte offset (aperture rule).
  const unsigned ldsAddr = (unsigned)(uintptr_t)(&zbuf[lane * 4u]);

  const unsigned long long fullWaves = n_chunks / CHUNKS_PER_WAVE;
  const unsigned long long W =
      (unsigned long long)blockIdx.x * WAVES_PER_BLOCK + (tid >> 5);

  if (W < fullWaves) {
    const unsigned long long addr = (unsigned long long)(uintptr_t)out +
                                    (W << 13) +               // 8KB per wave
                                    (unsigned long long)lane * 16ull;
    asm volatile(
        "s_clause 0xf\n\t"
        "global_store_async_from_lds_b128 %0, %1, off th:TH_STORE_NT\n\t"
        "global_store_async_from_lds_b128 %0, %1, off offset:512 th:TH_STORE_NT\n\t"
        "global_store_async_from_lds_b128 %0, %1, off offset:1024 th:TH_STORE_NT\n\t"
        "global_store_async_from_lds_b128 %0, %1, off offset:1536 th:TH_STORE_NT\n\t"
        "global_store_async_from_lds_b128 %0, %1, off offset:2048 th:TH_STORE_NT\n\t"
        "global_store_async_from_lds_b128 %0, %1, off offset:2560 th:TH_STORE_NT\n\t"
        "global_store_async_from_lds_b128 %0, %1, off offset:3072 th:TH_STORE_NT\n\t"
        "global_store_async_from_lds_b128 %0, %1, off offset:3584 th:TH_STORE_NT\n\t"
        "global_store_async_from_lds_b128 %0, %1, off offset:4096 th:TH_STORE_NT\n\t"
        "global_store_async_from_lds_b128 %0, %1, off offset:4608 th:TH_STORE_NT\n\t"
        "global_store_async_from_lds_b128 %0, %1, off offset:5120 th:TH_STORE_NT\n\t"
        "global_store_async_from_lds_b128 %0, %1, off offset:5632 th:TH_STORE_NT\n\t"
        "global_store_async_from_lds_b128 %0, %1, off offset:6144 th:TH_STORE_NT\n\t"
        "global_store_async_from_lds_b128 %0, %1, off offset:6656 th:TH_STORE_NT\n\t"
        "global_store_async_from_lds_b128 %0, %1, off offset:7168 th:TH_STORE_NT\n\t"
        "global_store_async_from_lds_b128 %0, %1, off offset:7680 th:TH_STORE_NT"
        :
        : "v"(addr), "v"(ldsAddr)
        : "memory");
  }

  // Tail A: leftover 16B chunks past the last full wave-region (none when
  // n_chunks % 512 == 0; for N=12288 the split is exact).
  const unsigned long long tailStart = fullWaves * CHUNKS_PER_WAVE;
  if (tailStart < n_chunks) {
    const unsigned long long stride =
        (unsigned long long)gridDim.x * (unsigned long long)blockDim.x;
    for (unsigned long long c =
             tailStart + (unsigned long long)blockIdx.x * blockDim.x + tid;
         c < n_chunks; c += stride) {
      const unsigned long long a =
          (unsigned long long)(uintptr_t)out + (c << 4);
      asm volatile("global_store_async_from_lds_b128 %0, %1, off th:TH_STORE_NT"
                   :
                   : "v"(a), "v"(ldsAddr)
                   : "memory");
    }
  }

  // Tail B: non-multiple-of-4 element tail (none for N=12288).
  if (blockIdx.x == 0 && tid < n_tail) {
    out[total_elems - 1ull - (unsigned long long)tid] = 0.0f;
  }

  // Explicitly drain the async store counter (S_ENDPGM would also wait-idle).
  asm volatile("s_wait_asynccnt 0" ::: "memory");
}

// ---------------------------------------------------------------------------
// Kernel 2: compute d[i] = -p*r + ics + t*vcsw + vc and scatter to diagonal.
// Runs after the fill in stream order; overwrites the zeroed diagonal slots.
// ---------------------------------------------------------------------------
__global__ void diag_kernel(const float* __restrict__ params,
                            const float* __restrict__ triggers,
                            const float* __restrict__ r_mask,
                            const float* __restrict__ ics_mask,
                            const float* __restrict__ vcsw_mask,
                            const float* __restrict__ vc_mask,
                            float* __restrict__ out, int n) {
  const int i = blockIdx.x * blockDim.x + threadIdx.x;
  if (i < n) {
    const float d = -params[i] * r_mask[i] + ics_mask[i] +
                    triggers[i] * vcsw_mask[i] + vc_mask[i];
    out[(unsigned long long)i * (unsigned long long)n + (unsigned long long)i] = d;
  }
}

extern "C" void kernel_launch(void* const* d_in, const int* in_sizes, int n_in,
                              void* d_out, int out_size, void* d_ws, size_t ws_size,
                              hipStream_t stream) {
  const float* params    = (const float*)d_in[0];
  const float* triggers  = (const float*)d_in[1];
  const float* r_mask    = (const float*)d_in[2];
  const float* ics_mask  = (const float*)d_in[3];
  const float* vcsw_mask = (const float*)d_in[4];
  const float* vc_mask   = (const float*)d_in[5];
  float* out = (float*)d_out;
  const int n = in_sizes[0];

  const unsigned long long total  = (unsigned long long)out_size;  // n*n
  const unsigned long long chunks = total >> 2;                    // 16B chunks
  const unsigned int tail         = (unsigned int)(total & 3ull);

  // One 8KB wave-region per wave; 8 waves per block.
  const unsigned long long waveRegions =
      (chunks + CHUNKS_PER_WAVE - 1ull) / CHUNKS_PER_WAVE;
  unsigned long long blocks64 =
      (waveRegions + WAVES_PER_BLOCK - 1ull) / WAVES_PER_BLOCK;
  if (blocks64 == 0) blocks64 = 1;
  const unsigned int blocks = (unsigned int)blocks64;  // 9216 for N=12288

  zero_fill_async_kernel<<<blocks, TPB, 0, stream>>>(out, chunks, tail, total);

  const unsigned int dblocks = (unsigned int)((n + TPB - 1) / TPB);
  diag_kernel<<<dblocks, TPB, 0, stream>>>(params, triggers, r_mask, ics_mask,
                                           vcsw_mask, vc_mask, out, n);
}